// NNCombFilterClone_28226525070331
// MI455X (gfx1250) — compile-verified
//
#include <hip/hip_runtime.h>

typedef float v2f __attribute__((ext_vector_type(2)));
typedef float v8f __attribute__((ext_vector_type(8)));

#define NTOTAL 16777216
#define DCH    4096      // channels == delay length
#define NBLK   4096      // NTOTAL / DCH
#define LGRP   16        // blocks per group (WMMA M dim)
#define NGRP   256       // NBLK / LGRP
#define NSUP   16        // supergroups (NGRP / 16)

// f^e for e in [0,15], branch-free binary exponentiation (no private arrays)
__device__ __forceinline__ float fpow16(float f, int e) {
    float r = 1.f, b = f;
#pragma unroll
    for (int bit = 0; bit < 4; ++bit) {
        r = (e & (1 << bit)) ? r * b : r;
        b = b * b;
    }
    return r;
}

__device__ __forceinline__ float pow2k(float f, int sq) {
    // f^(2^sq)
    float r = f;
    for (int i = 0; i < sq; ++i) r = r * r;
    return r;
}

// ---------------- Phase 1: group reductions A[g][d] = sum_t f^(15-t) x[g*16+t][d]
__global__ void comb_phase1(const float* __restrict__ x, const float* __restrict__ fb,
                            float* __restrict__ A) {
    const int tid = blockIdx.x * blockDim.x + threadIdx.x;   // g*DCH + d
    const int d = tid & (DCH - 1);
    const int g = tid >> 12;
    const float f = fb[0];
    const float* xp = x + (size_t)g * LGRP * DCH + d;
    float a = 0.f;
#pragma unroll
    for (int t = 0; t < LGRP; ++t)
        a = fmaf(f, a, xp[(size_t)t * DCH]);
    A[tid] = a;
}

// ---------------- Phase 2a: supergroup reductions AA[gg][d] over 16 groups (factor f^16)
__global__ void comb_phase2a(const float* __restrict__ A, const float* __restrict__ fb,
                             float* __restrict__ AA) {
    const int tid = blockIdx.x * blockDim.x + threadIdx.x;   // gg*DCH + d
    const int d = tid & (DCH - 1);
    const int gg = tid >> 12;
    const float fL = pow2k(fb[0], 4);                        // f^16
    const float* Ap = A + (size_t)gg * 16 * DCH + d;
    float a = 0.f;
#pragma unroll
    for (int q = 0; q < 16; ++q)
        a = fmaf(fL, a, Ap[(size_t)q * DCH]);
    AA[tid] = a;
}

// ---------------- Phase 2b: top-level serial scan over 16 supergroups (factor f^256)
__global__ void comb_phase2b(const float* __restrict__ AA, const float* __restrict__ fb,
                             float* __restrict__ CC) {
    const int d = blockIdx.x * blockDim.x + threadIdx.x;     // 4096 threads
    const float fLL = pow2k(fb[0], 8);                       // f^256
    float cc = 0.f;
#pragma unroll
    for (int gg = 0; gg < NSUP; ++gg) {
        CC[(size_t)gg * DCH + d] = cc;
        cc = fmaf(fLL, cc, AA[(size_t)gg * DCH + d]);
    }
}

// ---------------- Phase 2c: expand carries to every group: Cbuf[g][d]
__global__ void comb_phase2c(const float* __restrict__ A, const float* __restrict__ CC,
                             const float* __restrict__ fb, float* __restrict__ Cbuf) {
    const int tid = blockIdx.x * blockDim.x + threadIdx.x;   // gg*DCH + d
    const int d = tid & (DCH - 1);
    const int gg = tid >> 12;
    const float fL = pow2k(fb[0], 4);                        // f^16
    float cv = CC[(size_t)gg * DCH + d];
    const float* Ap = A + (size_t)gg * 16 * DCH + d;
    float* Cp = Cbuf + (size_t)gg * 16 * DCH + d;
#pragma unroll
    for (int q = 0; q < 16; ++q) {
        Cp[(size_t)q * DCH] = cv;
        cv = fmaf(fL, cv, Ap[(size_t)q * DCH]);
    }
}

// ---------------- Phase 3: WMMA reconstruction  D = T(16x16,tri) * X(16x16) + f^M*carry
// One wave per (group g, 64-channel tile); 4 chains of 4x V_WMMA_F32_16X16X4_F32.
__global__ void comb_phase3(const float* __restrict__ x, const float* __restrict__ fb,
                            const float* __restrict__ Cbuf, float* __restrict__ y) {
    const int lane = threadIdx.x & 31;
    const int wid  = blockIdx.x * (blockDim.x >> 5) + (threadIdx.x >> 5);
    const int g    = wid >> 6;            // 0..255
    const int d0   = (wid & 63) << 6;     // 64-channel tile base

    const float f = fb[0];
    const int h = lane >> 4;              // lane half (K/M offset selector)
    const int m = lane & 15;              // M for A, N for B/C/D

    // A fragments: element j of chunk kk holds T[m][K], K = 4*kk + j + 2*h,
    // T[i][t] = f^(i-1-t) for t < i else 0.
    v2f afrag[4];
#pragma unroll
    for (int kk = 0; kk < 4; ++kk) {
#pragma unroll
        for (int j = 0; j < 2; ++j) {
            const int t = 4 * kk + j + 2 * h;
            afrag[kk][j] = (t < m) ? fpow16(f, m - 1 - t) : 0.f;
        }
    }

    const size_t rowbase = (size_t)g * LGRP * DCH;
#pragma unroll
    for (int ct = 0; ct < 4; ++ct) {
        const int d0c = d0 + ct * 16;
        // carry for this channel (lane halves replicate the 16 channels)
        const float cv = Cbuf[(size_t)g * DCH + d0c + m];
        // C init: C[M][N] = f^M * carry[N], M = v + 8*h
        v8f acc;
#pragma unroll
        for (int v = 0; v < 8; ++v)
            acc[v] = fpow16(f, v + 8 * h) * cv;
        // chain 4 WMMAs over K
#pragma unroll
        for (int kk = 0; kk < 4; ++kk) {
            v2f bfrag;
#pragma unroll
            for (int j = 0; j < 2; ++j) {
                const int t = 4 * kk + j + 2 * h;
                bfrag[j] = x[rowbase + (size_t)t * DCH + d0c + m];
            }
            acc = __builtin_amdgcn_wmma_f32_16x16x4_f32(
                false, afrag[kk], false, bfrag, (short)0, acc, false, false);
        }
        // store D: row M = v + 8*h, col = d0c + m
#pragma unroll
        for (int v = 0; v < 8; ++v)
            y[rowbase + (size_t)(v + 8 * h) * DCH + d0c + m] = acc[v];
    }
}

extern "C" void kernel_launch(void* const* d_in, const int* in_sizes, int n_in,
                              void* d_out, int out_size, void* d_ws, size_t ws_size,
                              hipStream_t stream) {
    const float* x  = (const float*)d_in[0];
    const float* fb = (const float*)d_in[1];   // feedback scalar (1 elem)
    // d_in[2] = delay (int) == 4096, compile-time constant here
    float* y  = (float*)d_out;
    float* ws = (float*)d_ws;

    float* A    = ws;                                   // NGRP*DCH   (4 MB)
    float* Cb   = ws + (size_t)NGRP * DCH;              // NGRP*DCH   (4 MB)
    float* AA   = ws + (size_t)2 * NGRP * DCH;          // NSUP*DCH   (256 KB)
    float* CC   = ws + (size_t)2 * NGRP * DCH + (size_t)NSUP * DCH; // (256 KB)

    comb_phase1 <<<(NGRP * DCH) / 256, 256, 0, stream>>>(x, fb, A);
    comb_phase2a<<<(NSUP * DCH) / 256, 256, 0, stream>>>(A, fb, AA);
    comb_phase2b<<<DCH / 256,           256, 0, stream>>>(AA, fb, CC);
    comb_phase2c<<<(NSUP * DCH) / 256, 256, 0, stream>>>(A, CC, fb, Cb);
    // 16384 waves = 256 groups x 64 channel-tiles; 8 waves/block
    comb_phase3 <<<2048,                256, 0, stream>>>(x, fb, Cb, y);
}